// FullyConnectedThermalGNN_65721589563841
// MI455X (gfx1250) — compile-verified
//
#include <hip/hip_runtime.h>
#include <math.h>

#define NN 512
#define HH 64
#define LL 4
#define EPSL 1e-05f
#define NEG_BIG -3.402823466e38f

#define SPLIT 8            // src-dimension slices per dst (parallelism)
#define SLICE (NN / SPLIT) // 64 src per slice
#define PFD 16             // edge_attr prefetch distance (iterations)

typedef float v2f __attribute__((ext_vector_type(2)));
typedef float v8f __attribute__((ext_vector_type(8)));

// ---------------------------------------------------------------------------
// WMMA f32 GEMM: Out(512x64) = act( A(512x64) @ W(64x64) + bias + Cin )
// One wave per 16x16 output tile, grid = (32, 4). Uses V_WMMA_F32_16X16X4_F32.
// A-layout per lane: lanes 0-15 -> M=lane, K=k0..k0+1 ; lanes 16-31 -> M=lane-16, K=k0+2..k0+3
// B-layout mirrored; C/D: VGPR v -> row v (lanes<16) / v+8 (lanes>=16), col = lane&15.
// ---------------------------------------------------------------------------
__global__ __launch_bounds__(32) void wmma_gemm_512x64(
    const float* __restrict__ A, const float* __restrict__ W,
    const float* __restrict__ bias, const float* __restrict__ Cin,
    float* __restrict__ Out, int relu_act)
{
    const int mt   = blockIdx.x;          // 0..31
    const int nt   = blockIdx.y;          // 0..3
    const int lane = threadIdx.x;         // 0..31
    const int half = lane >> 4;           // 0/1
    const int l15  = lane & 15;
    const int row0 = mt * 16;
    const int col0 = nt * 16;

    v8f acc;
    if (Cin) {
#pragma unroll
        for (int v = 0; v < 8; ++v)
            acc[v] = Cin[(row0 + v + half * 8) * HH + col0 + l15];
    } else {
#pragma unroll
        for (int v = 0; v < 8; ++v) acc[v] = 0.0f;
    }

#pragma unroll
    for (int k0 = 0; k0 < HH; k0 += 4) {
        const int ka = k0 + half * 2;
        v2f a, b;
        a.x = A[(row0 + l15) * HH + ka];
        a.y = A[(row0 + l15) * HH + ka + 1];
        b.x = W[(ka)     * HH + col0 + l15];
        b.y = W[(ka + 1) * HH + col0 + l15];
        acc = __builtin_amdgcn_wmma_f32_16x16x4_f32(
                  false, a, false, b, (short)0, acc, false, false);
    }

    const float bc = bias ? bias[col0 + l15] : 0.0f;
#pragma unroll
    for (int v = 0; v < 8; ++v) {
        float val = acc[v] + bc;
        if (relu_act) val = fmaxf(val, 0.0f);
        Out[(row0 + v + half * 8) * HH + col0 + l15] = val;
    }
}

// ---------------------------------------------------------------------------
// Encoder: h = relu(x @ w1 + b1) @ w2 + b2   (x: 512x6)
// ---------------------------------------------------------------------------
__global__ __launch_bounds__(64) void enc_kernel(
    const float* __restrict__ x, const float* __restrict__ w1,
    const float* __restrict__ b1, const float* __restrict__ w2,
    const float* __restrict__ b2, float* __restrict__ h)
{
    const int row = blockIdx.x, c = threadIdx.x;
    float t = b1[c];
#pragma unroll
    for (int k = 0; k < 6; ++k) t += x[row * 6 + k] * w1[k * HH + c];
    t = fmaxf(t, 0.0f);
    __shared__ float sm[HH];
    sm[c] = t;
    __syncthreads();
    float o = b2[c];
#pragma unroll 8
    for (int k = 0; k < HH; ++k) o += sm[k] * w2[k * HH + c];
    h[row * HH + c] = o;
}

// ---------------------------------------------------------------------------
// Fused per-edge pass: attention logits + online softmax + weighted value acc.
// Block = 1 wave32.  blockIdx.x = dst node j (uniform), blockIdx.y = src slice.
// Lane owns channels c0=2*lane, c1=2*lane+1. Partials: [m, ssum, acc[64]] (66 f)
// edge_attr is the only cold stream (12KB stride per iter) -> explicit prefetch.
// ---------------------------------------------------------------------------
__global__ __launch_bounds__(32) void edge_kernel(
    const float* __restrict__ Pd, const float* __restrict__ Ps,
    const float* __restrict__ Qs, const float* __restrict__ ea,
    const float* __restrict__ Wea, const float* __restrict__ Vea,
    const float* __restrict__ w2, const float* __restrict__ b2p,
    float* __restrict__ part)
{
    const int j    = blockIdx.x;
    const int s    = blockIdx.y;
    const int lane = threadIdx.x;
    const int c0   = 2 * lane, c1 = 2 * lane + 1;

    const float pd0 = Pd[j * HH + c0], pd1 = Pd[j * HH + c1];
    const float w20 = w2[c0],          w21 = w2[c1];
    const float b2  = b2p[0];

    float wea0[6], wea1[6], vea0[6], vea1[6];
#pragma unroll
    for (int k = 0; k < 6; ++k) {
        wea0[k] = Wea[k * HH + c0]; wea1[k] = Wea[k * HH + c1];
        vea0[k] = Vea[k * HH + c0]; vea1[k] = Vea[k * HH + c1];
    }

    float m = NEG_BIG, ssum = 0.0f, acc0 = 0.0f, acc1 = 0.0f;
    const int i0 = s * SLICE;

#pragma unroll 2
    for (int ii = 0; ii < SLICE; ++ii) {
        const int i = i0 + ii;
        const float2 ps = *(const float2*)(Ps + (size_t)i * HH + c0);
        const float2 qs = *(const float2*)(Qs + (size_t)i * HH + c0);
        const float* eaP = ea + ((size_t)i * NN + j) * 6;
        float e0 = eaP[0], e1 = eaP[1], e2 = eaP[2],
              e3 = eaP[3], e4 = eaP[4], e5 = eaP[5];

        // prefetch edge_attr row PFD iterations ahead (gfx1250 global_prefetch)
        {
            int ip = i + PFD; if (ip > NN - 1) ip = NN - 1;
            __builtin_prefetch(ea + ((size_t)ip * NN + j) * 6, 0, 0);
        }

        // attention pre-activation (per-channel)
        float s0 = pd0 + ps.x, s1 = pd1 + ps.y;
        s0 += e0*wea0[0] + e1*wea0[1] + e2*wea0[2] + e3*wea0[3] + e4*wea0[4] + e5*wea0[5];
        s1 += e0*wea1[0] + e1*wea1[1] + e2*wea1[2] + e3*wea1[3] + e4*wea1[4] + e5*wea1[5];
        s0 = (s0 > 0.0f) ? s0 : 0.2f * s0;           // leaky_relu(0.2)
        s1 = (s1 > 0.0f) ? s1 : 0.2f * s1;

        float dotp = s0 * w20 + s1 * w21;            // dot with att_w2
#pragma unroll
        for (int off = 16; off > 0; off >>= 1)
            dotp += __shfl_xor(dotp, off, 32);
        const float a = dotp + b2;                   // logit (uniform on wave)

        // value hidden (val_w2 applied after aggregation — linearity)
        float t0 = qs.x, t1 = qs.y;
        t0 += e0*vea0[0] + e1*vea0[1] + e2*vea0[2] + e3*vea0[3] + e4*vea0[4] + e5*vea0[5];
        t1 += e0*vea1[0] + e1*vea1[1] + e2*vea1[2] + e3*vea1[3] + e4*vea1[4] + e5*vea1[5];
        t0 = fmaxf(t0, 0.0f); t1 = fmaxf(t1, 0.0f);

        // online softmax accumulate
        const float mn = fmaxf(m, a);
        const float sc = __expf(m - mn);
        const float w  = __expf(a - mn);
        ssum = ssum * sc + w;
        acc0 = acc0 * sc + w * t0;
        acc1 = acc1 * sc + w * t1;
        m = mn;
    }

    float* p = part + ((size_t)j * SPLIT + s) * 66;
    if (lane == 0) { p[0] = m; p[1] = ssum; }
    p[2 + c0] = acc0;
    p[2 + c1] = acc1;
}

// ---------------------------------------------------------------------------
// Merge SPLIT online-softmax partials per dst -> aggraw (pre-val_w2 aggregation)
// ---------------------------------------------------------------------------
__global__ __launch_bounds__(64) void combine_kernel(
    const float* __restrict__ part, float* __restrict__ aggraw)
{
    const int j = blockIdx.x, c = threadIdx.x;
    float M = NEG_BIG;
#pragma unroll
    for (int s = 0; s < SPLIT; ++s)
        M = fmaxf(M, part[((size_t)j * SPLIT + s) * 66]);
    float tot = 0.0f, accv = 0.0f;
#pragma unroll
    for (int s = 0; s < SPLIT; ++s) {
        const float* p = part + ((size_t)j * SPLIT + s) * 66;
        const float w = __expf(p[0] - M);
        tot  += p[1] * w;
        accv += p[2 + c] * w;
    }
    aggraw[j * HH + c] = accv / tot;
}

// ---------------------------------------------------------------------------
// LayerNorm over H=64: h = g*(r-mu)/sqrt(var+eps) + b
// ---------------------------------------------------------------------------
__global__ __launch_bounds__(64) void ln_kernel(
    const float* __restrict__ r, const float* __restrict__ g,
    const float* __restrict__ b, float* __restrict__ h)
{
    const int row = blockIdx.x, c = threadIdx.x;
    __shared__ float sm[HH];
    const float v = r[row * HH + c];
    sm[c] = v; __syncthreads();
#pragma unroll
    for (int off = 32; off > 0; off >>= 1) {
        if (c < off) sm[c] += sm[c + off];
        __syncthreads();
    }
    const float mu = sm[0] * (1.0f / HH);
    __syncthreads();
    const float d = v - mu;
    sm[c] = d * d; __syncthreads();
#pragma unroll
    for (int off = 32; off > 0; off >>= 1) {
        if (c < off) sm[c] += sm[c + off];
        __syncthreads();
    }
    const float var = sm[0] * (1.0f / HH);
    h[row * HH + c] = g[c] * d / sqrtf(var + EPSL) + b[c];
}

// ---------------------------------------------------------------------------
// Decoder: out = relu(h @ w1 + b1) @ w2 + b2   (64 -> 1)
// ---------------------------------------------------------------------------
__global__ __launch_bounds__(64) void dec_kernel(
    const float* __restrict__ h, const float* __restrict__ w1,
    const float* __restrict__ b1, const float* __restrict__ w2,
    const float* __restrict__ b2, float* __restrict__ out)
{
    const int row = blockIdx.x, c = threadIdx.x;
    float t = b1[c];
    const float* hr = h + row * HH;
#pragma unroll 8
    for (int k = 0; k < HH; ++k) t += hr[k] * w1[k * HH + c];
    t = fmaxf(t, 0.0f);
    __shared__ float sm[HH];
    sm[c] = t * w2[c];
    __syncthreads();
#pragma unroll
    for (int off = 32; off > 0; off >>= 1) {
        if (c < off) sm[c] += sm[c + off];
        __syncthreads();
    }
    if (c == 0) out[row] = sm[0] + b2[0];
}

// ---------------------------------------------------------------------------
extern "C" void kernel_launch(void* const* d_in, const int* in_sizes, int n_in,
                              void* d_out, int out_size, void* d_ws, size_t ws_size,
                              hipStream_t stream)
{
    (void)in_sizes; (void)n_in; (void)out_size; (void)ws_size;
    const float* x         = (const float*)d_in[0];
    const float* edge_attr = (const float*)d_in[1];
    const float* enc_w1    = (const float*)d_in[2];
    const float* enc_b1    = (const float*)d_in[3];
    const float* enc_w2    = (const float*)d_in[4];
    const float* enc_b2    = (const float*)d_in[5];
    const float* att_w1    = (const float*)d_in[6];   // (L,134,64)
    const float* att_b1    = (const float*)d_in[7];   // (L,64)
    const float* att_w2    = (const float*)d_in[8];   // (L,64,1)
    const float* att_b2    = (const float*)d_in[9];   // (L,1)
    const float* val_w1    = (const float*)d_in[10];  // (L,70,64)
    const float* val_b1    = (const float*)d_in[11];  // (L,64)
    const float* val_w2    = (const float*)d_in[12];  // (L,64,64)
    const float* val_b2    = (const float*)d_in[13];  // (L,64)
    const float* upd_w1    = (const float*)d_in[14];  // (L,128,64)
    const float* upd_b1    = (const float*)d_in[15];  // (L,64)
    const float* upd_w2    = (const float*)d_in[16];  // (L,64,64)
    const float* upd_b2    = (const float*)d_in[17];  // (L,64)
    const float* ln_g      = (const float*)d_in[18];  // (L,64)
    const float* ln_b      = (const float*)d_in[19];  // (L,64)
    const float* dec_w1    = (const float*)d_in[20];
    const float* dec_b1    = (const float*)d_in[21];
    const float* dec_w2    = (const float*)d_in[22];
    const float* dec_b2    = (const float*)d_in[23];
    float* out = (float*)d_out;

    float* f = (float*)d_ws;
    const size_t NH = (size_t)NN * HH;      // 32768
    float* h      = f;
    float* Pd     = f + 1 * NH;
    float* Ps     = f + 2 * NH;
    float* Qs     = f + 3 * NH;
    float* aggraw = f + 4 * NH;
    float* agg    = f + 5 * NH;
    float* tmp    = f + 6 * NH;
    float* z      = f + 7 * NH;
    float* r      = f + 8 * NH;
    float* part   = f + 9 * NH;             // 512*SPLIT*66 floats

    enc_kernel<<<NN, HH, 0, stream>>>(x, enc_w1, enc_b1, enc_w2, enc_b2, h);

    const dim3 gG(32, 4);          // WMMA grid: 32 M-tiles x 4 N-tiles
    const dim3 gE(NN, SPLIT);      // edge grid: dst x src-slice

    for (int l = 0; l < LL; ++l) {
        const float* aw1 = att_w1 + (size_t)l * 134 * HH;
        const float* vw1 = val_w1 + (size_t)l * 70 * HH;
        const float* uw1 = upd_w1 + (size_t)l * 128 * HH;

        // Node-level pre-projections (WMMA f32)
        wmma_gemm_512x64<<<gG, 32, 0, stream>>>(h, aw1,            nullptr,         nullptr, Pd, 0);
        wmma_gemm_512x64<<<gG, 32, 0, stream>>>(h, aw1 + 64 * HH,  att_b1 + l * HH, nullptr, Ps, 0);
        wmma_gemm_512x64<<<gG, 32, 0, stream>>>(h, vw1,            val_b1 + l * HH, nullptr, Qs, 0);

        // Fused per-edge attention + online softmax + value aggregation
        edge_kernel<<<gE, 32, 0, stream>>>(Pd, Ps, Qs, edge_attr,
                                           aw1 + 128 * HH, vw1 + 64 * HH,
                                           att_w2 + (size_t)l * HH, att_b2 + l, part);
        combine_kernel<<<NN, HH, 0, stream>>>(part, aggraw);

        // agg = aggraw @ val_w2 + val_b2   (val_w2 pulled out of the edge loop)
        wmma_gemm_512x64<<<gG, 32, 0, stream>>>(aggraw, val_w2 + (size_t)l * HH * HH,
                                                val_b2 + l * HH, nullptr, agg, 0);

        // update MLP with fused concat-split + residual via WMMA C operand
        wmma_gemm_512x64<<<gG, 32, 0, stream>>>(h,   uw1,           nullptr,         nullptr, tmp, 0);
        wmma_gemm_512x64<<<gG, 32, 0, stream>>>(agg, uw1 + 64 * HH, upd_b1 + l * HH, tmp,     z,   1);
        wmma_gemm_512x64<<<gG, 32, 0, stream>>>(z,   upd_w2 + (size_t)l * HH * HH,
                                                upd_b2 + l * HH, h, r, 0);  // r = u + h

        ln_kernel<<<NN, HH, 0, stream>>>(r, ln_g + l * HH, ln_b + l * HH, h);
    }

    dec_kernel<<<NN, HH, 0, stream>>>(h, dec_w1, dec_b1, dec_w2, dec_b2, out);
}